// TAM_19026705122089
// MI455X (gfx1250) — compile-verified
//
#include <hip/hip_runtime.h>
#include <hip/hip_bf16.h>

// ---- problem constants (from reference) ----
#define NN 8
#define CC 256
#define TT 8
#define T2 16      // 2*t
#define CR 64      // c/4
#define KK 3
#define HH 56
#define WW 56
#define HW (HH*WW)         // 3136
#define HW4 (HW/4)         // 784
#define NC (NN*CC)         // 2048
#define EPS 1e-5f

typedef __attribute__((ext_vector_type(2))) float v2f;
typedef __attribute__((ext_vector_type(8))) float v8f;

// ============================================================
// Kernel 1: spatial mean pool.  grid = N*C*T blocks, 256 thr.
// Each block reduces one 56x56 plane (784 float4 loads).
// ============================================================
__global__ __launch_bounds__(256)
void tam_pool_kernel(const float* __restrict__ x, float* __restrict__ pool) {
    const int b = blockIdx.x;                 // (n*C + c)*T + t
    const float4* xp = (const float4*)(x + (size_t)b * HW);
    float s = 0.f;
    for (int i = threadIdx.x; i < HW4; i += 256) {
        float4 v = xp[i];
        s += (v.x + v.y) + (v.z + v.w);
    }
    __shared__ float red[256];
    red[threadIdx.x] = s;
    __syncthreads();
    #pragma unroll
    for (int off = 128; off > 0; off >>= 1) {
        if (threadIdx.x < off) red[threadIdx.x] += red[threadIdx.x + off];
        __syncthreads();
    }
    if (threadIdx.x == 0) pool[b] = red[0] * (1.0f / (float)HW);
}

// ============================================================
// Kernel 2: G-branch via V_WMMA_F32_16X16X4_F32.
//   y[2048x16] = pool[2048x8] @ wg1^T[8x16]  (two K=4 WMMA steps)
//   BN+ReLU in accumulator, then 16x3 projection + softmax.
// grid = NC/(16 rows * 8 waves) = 16 blocks of 256 threads.
// A layout (16x4 f32): lane L<16 -> row L, K={0,1} in v0,v1;
//                      lane L>=16 -> row L-16, K={2,3}.
// B layout (4x16 f32): same K split, N = lane&15.
// C layout (16x16 f32): vgpr r -> M = r + 8*(lane>>4), N = lane&15.
// ============================================================
__global__ __launch_bounds__(256)
void tam_gbranch_kernel(const float* __restrict__ pool,
                        const float* __restrict__ wg1,     // [16][8]
                        const float* __restrict__ g_gamma, // [16]
                        const float* __restrict__ g_beta,
                        const float* __restrict__ g_mean,
                        const float* __restrict__ g_var,
                        const float* __restrict__ wg2,     // [3][16]
                        float* __restrict__ kern_out) {    // [NC][3]
    const int lane  = threadIdx.x & 31;
    const int wave  = threadIdx.x >> 5;
    const int rowbase = (blockIdx.x * 8 + wave) * 16;
    const int col   = lane & 15;                 // N index (and A row index)
    const int khalf = (lane >> 4) << 1;          // 0 or 2

    // A: pool rows (K-chunks 0..3 and 4..7)
    const float* prow = pool + (size_t)(rowbase + col) * TT;
    v2f a0, a1, b0, b1;
    a0.x = prow[khalf];     a0.y = prow[khalf + 1];
    a1.x = prow[khalf + 4]; a1.y = prow[khalf + 5];
    // B[k][j] = wg1[j][k]
    const float* wrow = wg1 + (size_t)col * TT;
    b0.x = wrow[khalf];     b0.y = wrow[khalf + 1];
    b1.x = wrow[khalf + 4]; b1.y = wrow[khalf + 5];

    v8f acc = {};
    acc = __builtin_amdgcn_wmma_f32_16x16x4_f32(false, a0, false, b0,
                                                (short)0, acc, false, false);
    acc = __builtin_amdgcn_wmma_f32_16x16x4_f32(false, a1, false, b1,
                                                (short)0, acc, false, false);

    // BN (eval) + ReLU: column constants live in lane (N = col for all 8 vgprs)
    const float sc = g_gamma[col] * __frsqrt_rn(g_var[col] + EPS);
    const float sh = g_beta[col] - g_mean[col] * sc;

    __shared__ float ylds[8][16][17];
    #pragma unroll
    for (int r = 0; r < 8; ++r) {
        float v = acc[r] * sc + sh;
        v = v > 0.f ? v : 0.f;
        const int m = r + ((lane >> 4) << 3);
        ylds[wave][m][col] = v;
    }
    __syncthreads();

    // z = y @ wg2^T (16x3), softmax over 3. One row per lane (lanes 0..15).
    if (lane < 16) {
        const float* yr = ylds[wave][lane];
        float z[3];
        #pragma unroll
        for (int p = 0; p < 3; ++p) {
            float s = 0.f;
            #pragma unroll
            for (int j = 0; j < 16; ++j) s += yr[j] * wg2[p * 16 + j];
            z[p] = s;
        }
        const float mx = fmaxf(z[0], fmaxf(z[1], z[2]));
        const float e0 = __expf(z[0] - mx);
        const float e1 = __expf(z[1] - mx);
        const float e2 = __expf(z[2] - mx);
        const float inv = 1.f / (e0 + e1 + e2);
        float* ko = kern_out + (size_t)(rowbase + lane) * 3;
        ko[0] = e0 * inv; ko[1] = e1 * inv; ko[2] = e2 * inv;
    }
}

// ============================================================
// Kernel 3: L-branch (tiny). grid = N blocks of 256 threads.
//   conv1d(pool, wl1) -> BN -> ReLU -> wl2 projection -> sigmoid
// ============================================================
__global__ __launch_bounds__(256)
void tam_lbranch_kernel(const float* __restrict__ pool,
                        const float* __restrict__ wl1,     // [64][256][3]
                        const float* __restrict__ l_gamma, // [64]
                        const float* __restrict__ l_beta,
                        const float* __restrict__ l_mean,
                        const float* __restrict__ l_var,
                        const float* __restrict__ wl2,     // [256][64]
                        float* __restrict__ loc_out) {     // [NC][T]
    const int n = blockIdx.x;
    __shared__ float pl[CC][TT + 2];   // zero-padded pool[n]
    __shared__ float locr[CR][TT];

    for (int i = threadIdx.x; i < CC; i += 256) {
        pl[i][0] = 0.f;
        pl[i][TT + 1] = 0.f;
        #pragma unroll
        for (int t = 0; t < TT; ++t)
            pl[i][t + 1] = pool[((size_t)n * CC + i) * TT + t];
    }
    __syncthreads();

    for (int idx = threadIdx.x; idx < CR * TT; idx += 256) {
        const int o  = idx >> 3;
        const int t0 = idx & 7;
        float s = 0.f;
        const float* wrow = wl1 + (size_t)o * CC * 3;
        for (int i = 0; i < CC; ++i) {
            s += wrow[i * 3 + 0] * pl[i][t0]
               + wrow[i * 3 + 1] * pl[i][t0 + 1]
               + wrow[i * 3 + 2] * pl[i][t0 + 2];
        }
        const float sc = l_gamma[o] * __frsqrt_rn(l_var[o] + EPS);
        float v = s * sc + (l_beta[o] - l_mean[o] * sc);
        locr[o][t0] = v > 0.f ? v : 0.f;
    }
    __syncthreads();

    const int c = threadIdx.x;        // blockDim == C == 256
    float acc[TT] = {};
    const float* w2r = wl2 + (size_t)c * CR;
    for (int o = 0; o < CR; ++o) {
        const float w = w2r[o];
        #pragma unroll
        for (int t0 = 0; t0 < TT; ++t0) acc[t0] += w * locr[o][t0];
    }
    #pragma unroll
    for (int t0 = 0; t0 < TT; ++t0) {
        loc_out[((size_t)n * CC + c) * TT + t0] = 1.f / (1.f + __expf(-acc[t0]));
    }
}

// ============================================================
// Kernel 4: fused gate + dynamic 3-tap temporal conv (BW-bound).
// grid = N*C blocks; each block owns all 8 time planes of one
// (n,c) channel. Per float4 position: 8 b128 loads, 8 b128 stores,
// x read exactly once here.
//   out[t0] = m0[t0]*x[t0-1] + m1[t0]*x[t0] + m2[t0]*x[t0+1]
//   m_j[t0] = kern[j] * sigmoid_gate[t0+j-1]   (0 outside [0,T))
// ============================================================
__global__ __launch_bounds__(256)
void tam_fuse_kernel(const float* __restrict__ x,
                     const float* __restrict__ kern,   // [NC][3]
                     const float* __restrict__ loc,    // [NC][T]
                     float* __restrict__ out) {
    const int ncid = blockIdx.x;
    const float k0 = kern[(size_t)ncid * 3 + 0];
    const float k1 = kern[(size_t)ncid * 3 + 1];
    const float k2 = kern[(size_t)ncid * 3 + 2];
    float g[TT];
    #pragma unroll
    for (int t = 0; t < TT; ++t) g[t] = loc[(size_t)ncid * TT + t];

    float m0[TT], m1[TT], m2[TT];
    #pragma unroll
    for (int t0 = 0; t0 < TT; ++t0) {
        m0[t0] = (t0 - 1 >= 0) ? k0 * g[t0 - 1] : 0.f;
        m1[t0] = k1 * g[t0];
        m2[t0] = (t0 + 1 < TT) ? k2 * g[t0 + 1] : 0.f;
    }

    const float4* xb = (const float4*)(x + (size_t)ncid * TT * HW);
    float4*       ob = (float4*)(out + (size_t)ncid * TT * HW);

    for (int pos = threadIdx.x; pos < HW4; pos += 256) {
        float4 xv[TT];
        #pragma unroll
        for (int t = 0; t < TT; ++t) xv[t] = xb[(size_t)t * HW4 + pos];
        #pragma unroll
        for (int t0 = 0; t0 < TT; ++t0) {
            const float4 zero = make_float4(0.f, 0.f, 0.f, 0.f);
            const float4 lo = (t0 > 0)      ? xv[t0 - 1] : zero;
            const float4 mi = xv[t0];
            const float4 hi = (t0 < TT - 1) ? xv[t0 + 1] : zero;
            float4 r;
            r.x = m0[t0] * lo.x + m1[t0] * mi.x + m2[t0] * hi.x;
            r.y = m0[t0] * lo.y + m1[t0] * mi.y + m2[t0] * hi.y;
            r.z = m0[t0] * lo.z + m1[t0] * mi.z + m2[t0] * hi.z;
            r.w = m0[t0] * lo.w + m1[t0] * mi.w + m2[t0] * hi.w;
            ob[(size_t)t0 * HW4 + pos] = r;
        }
    }
}

// ============================================================
extern "C" void kernel_launch(void* const* d_in, const int* in_sizes, int n_in,
                              void* d_out, int out_size, void* d_ws, size_t ws_size,
                              hipStream_t stream) {
    const float* x       = (const float*)d_in[0];
    const float* wg1     = (const float*)d_in[1];
    const float* g_gamma = (const float*)d_in[2];
    const float* g_beta  = (const float*)d_in[3];
    const float* g_mean  = (const float*)d_in[4];
    const float* g_var   = (const float*)d_in[5];
    const float* wg2     = (const float*)d_in[6];
    const float* wl1     = (const float*)d_in[7];
    const float* l_gamma = (const float*)d_in[8];
    const float* l_beta  = (const float*)d_in[9];
    const float* l_mean  = (const float*)d_in[10];
    const float* l_var   = (const float*)d_in[11];
    const float* wl2     = (const float*)d_in[12];
    float* out = (float*)d_out;

    // workspace layout
    float* pool = (float*)d_ws;            // NC*T      = 16384 floats
    float* kern = pool + (size_t)NC * TT;  // NC*3      =  6144 floats
    float* loc  = kern + (size_t)NC * 3;   // NC*T      = 16384 floats

    tam_pool_kernel<<<NC * TT, 256, 0, stream>>>(x, pool);
    tam_gbranch_kernel<<<NC / 128, 256, 0, stream>>>(pool, wg1, g_gamma, g_beta,
                                                     g_mean, g_var, wg2, kern);
    tam_lbranch_kernel<<<NN, 256, 0, stream>>>(pool, wl1, l_gamma, l_beta,
                                               l_mean, l_var, wl2, loc);
    tam_fuse_kernel<<<NC, 256, 0, stream>>>(x, kern, loc, out);
}